// CapacityEstimation_88699664597268
// MI455X (gfx1250) — compile-verified
//
#include <hip/hip_runtime.h>
#include <stdint.h>

// ---------------------------------------------------------------------------
// CapacityEstimation pipeline for gfx1250 (MI455X), f32 end-to-end.
// Stage 1: SPP pyramid pooling (memory-bound, coalesced, dedup'd series)
// Stage 2: inception_a + cnn1 + cnn2 (async global->LDS row staging)
// Stage 3: feedforward layer1 as f32 WMMA GEMM, operands staged via TDM
// Stage 4: feedforward tail 64->16->4->1
// Stage 5: inception_b (32 x 96 -> 32)
// ---------------------------------------------------------------------------

#define LSEQ   2048
#define NBINS  91
#define BATCH  32
#define CELLS  96
#define NTOT   (BATCH * CELLS)   // 3072
#define KPAD   96                // K padded from 91 for WMMA
#define H1DIM  64

typedef float v2f __attribute__((ext_vector_type(2)));
typedef float v8f __attribute__((ext_vector_type(8)));
typedef unsigned int v4u __attribute__((ext_vector_type(4)));
typedef int v4i __attribute__((ext_vector_type(4)));
typedef int v8i __attribute__((ext_vector_type(8)));

#define AS1 __attribute__((address_space(1)))
#define AS3 __attribute__((address_space(3)))

#if __has_builtin(__builtin_amdgcn_global_load_async_to_lds_b32)
#define HAVE_ASYNC_LDS 1
#else
#define HAVE_ASYNC_LDS 0
#endif

#if __has_builtin(__builtin_amdgcn_tensor_load_to_lds) && __has_builtin(__builtin_amdgcn_s_wait_tensorcnt)
#define HAVE_TDM 1
#else
#define HAVE_TDM 0
#endif

__device__ __forceinline__ float relu_f(float x) { return x > 0.f ? x : 0.f; }

// generic LDS pointer -> 32-bit LDS byte offset (low 32 bits of flat address)
__device__ __forceinline__ unsigned lds_off(const void* p) {
  return (unsigned)(uintptr_t)p;
}

#if HAVE_ASYNC_LDS
__device__ __forceinline__ void async_cp_b32(void* lds, const void* glob) {
  // builtin signature (per hipcc diagnostic): (AS1 int*, AS3 int*, imm off, imm cpol)
  __builtin_amdgcn_global_load_async_to_lds_b32(
      (AS1 int*)(uintptr_t)glob,
      (AS3 int*)(unsigned)(uintptr_t)lds, 0, 0);
}
__device__ __forceinline__ void async_wait0() {
#if __has_builtin(__builtin_amdgcn_s_wait_asynccnt)
  __builtin_amdgcn_s_wait_asynccnt(0);
#else
  asm volatile("s_wait_asynccnt 0x0" ::: "memory");
#endif
  asm volatile("" ::: "memory");
}
#endif

#if HAVE_TDM
// Issue a 2-D TENSOR_LOAD_TO_LDS: tile (tile_d0 x tile_d1) of 4-byte elements
// from row-major tensor with row stride stride0 (elements) at gaddr, into
// contiguous LDS at lds_byte_off. D# bit layout per CDNA5 ISA ch.8.
__device__ __forceinline__ void tdm_load_2d(unsigned lds_byte_off, const void* gaddr,
                                            unsigned tensor_d0, unsigned tensor_d1,
                                            unsigned stride0,
                                            unsigned tile_d0, unsigned tile_d1) {
  unsigned long long ga = (unsigned long long)(uintptr_t)gaddr;
  v4u g0;
  g0.x = 1u;                                              // count=1 (valid user D#)
  g0.y = lds_byte_off;                                    // lds_addr [63:32]
  g0.z = (unsigned)(ga & 0xffffffffu);                    // global_addr[31:0]
  g0.w = (unsigned)((ga >> 32) & 0x01ffffffu) | (2u << 30); // ga[56:32] | type=2
  v8i g1;
  g1[0] = (int)(2u << 16);                                // data_size=2 (4 bytes)
  g1[1] = (int)((tensor_d0 & 0xffffu) << 16);             // tensor_dim0[15:0]
  g1[2] = (int)((tensor_d0 >> 16) | ((tensor_d1 & 0xffffu) << 16));
  g1[3] = (int)((tensor_d1 >> 16) | ((tile_d0 & 0xffffu) << 16));
  g1[4] = (int)(tile_d1 & 0xffffu);                       // tile_dim1; tile_dim2=0
  g1[5] = (int)stride0;                                   // tensor_dim0_stride[31:0]
  g1[6] = 0;
  g1[7] = 0;
  v4i gz = {0, 0, 0, 0};
#if __clang_major__ >= 23
  v8i gz8 = {0, 0, 0, 0, 0, 0, 0, 0};
  __builtin_amdgcn_tensor_load_to_lds(g0, g1, gz, gz, gz8, 0);
#else
  __builtin_amdgcn_tensor_load_to_lds(g0, g1, gz, gz, 0);
#endif
}
#endif

// adaptive avg-pool bin bounds: bin p of 91 -> level lvl, seg k, [s, e)
__device__ __forceinline__ void bin_bounds(int p, int* s, int* e) {
  int off = 0, lvl = 1;
  while (p >= off + lvl) { off += lvl; ++lvl; }
  int k = p - off;
  *s = (k * LSEQ) / lvl;
  *e = ((k + 1) * LSEQ + lvl - 1) / lvl;   // ceil
}

// ---------------------------------------------------------------------------
// Kernel 1: SPP for all distinct series.
// grid = (91, 32, 2), block = 96.
// ---------------------------------------------------------------------------
__global__ void k_spp(const float* __restrict__ T, const float* __restrict__ V,
                      const float* __restrict__ Tm, const float* __restrict__ I,
                      float* __restrict__ sppT, float* __restrict__ sppI,
                      float* __restrict__ sppV, float* __restrict__ sppTm) {
  const int p = blockIdx.x;
  const int b = blockIdx.y;
  const int z = blockIdx.z;
  const int c = threadIdx.x;
  int s, e;
  bin_bounds(p, &s, &e);
  const float inv = 1.0f / (float)(e - s);
  if (z == 0) {
    const float* base = V + ((size_t)b * LSEQ) * CELLS + c;
    float acc = 0.f;
    for (int t = s; t < e; ++t) acc += base[(size_t)t * CELLS];
    sppV[((size_t)(b * CELLS + c)) * NBINS + p] = acc * inv;
  } else {
    if (c < 48) {
      const float* base = Tm + ((size_t)b * LSEQ) * 48 + c;
      float acc = 0.f;
      for (int t = s; t < e; ++t) acc += base[(size_t)t * 48];
      sppTm[((size_t)(b * 48 + c)) * NBINS + p] = acc * inv;
    } else if (c == 48) {
      const float* base = T + (size_t)b * LSEQ;
      float acc = 0.f;
      for (int t = s; t < e; ++t) acc += base[t];
      sppT[b * NBINS + p] = acc * inv;
    } else if (c == 49) {
      const float* base = I + (size_t)b * LSEQ;
      float acc = 0.f;
      for (int t = s; t < e; ++t) acc += base[t];
      sppI[b * NBINS + p] = acc * inv;
    }
  }
}

// ---------------------------------------------------------------------------
// Kernel 2: pack W1 (64 x 91, [o][k]) into Bpad (96 x 64, [k][o]).
// ---------------------------------------------------------------------------
__global__ void k_prep_b(const float* __restrict__ W1, float* __restrict__ Bp) {
  int idx = blockIdx.x * blockDim.x + threadIdx.x;
  if (idx >= KPAD * H1DIM) return;
  int k = idx / H1DIM, o = idx % H1DIM;
  Bp[idx] = (k < NBINS) ? W1[o * NBINS + k] : 0.f;
}

// ---------------------------------------------------------------------------
// Kernel 3: inception_a + cnn1 + cnn2 per n=(b,cell). grid = 3072, block = 96.
// SPP rows staged into LDS via GLOBAL_LOAD_ASYNC_TO_LDS_B32 when available.
// ---------------------------------------------------------------------------
__global__ void k_incept_a(const float* __restrict__ sppT, const float* __restrict__ sppI,
                           const float* __restrict__ sppV, const float* __restrict__ sppTm,
                           const float* __restrict__ w1,  const float* __restrict__ b1,
                           const float* __restrict__ w2,  const float* __restrict__ b2,
                           const float* __restrict__ w3a, const float* __restrict__ b3a,
                           const float* __restrict__ w3b, const float* __restrict__ b3b,
                           const float* __restrict__ c1w, const float* __restrict__ c1b,
                           const float* __restrict__ c2w, const float* __restrict__ c2b,
                           float* __restrict__ Z) {
  __shared__ float sx[3][NBINS];   // channels T, V, I
  __shared__ float stm[NBINS];
  __shared__ float sh[5][NBINS];   // branch3 hidden
  const int n = blockIdx.x;
  const int b = n / CELLS, c = n % CELLS;
  const int p = threadIdx.x;
  if (p < NBINS) {
#if HAVE_ASYNC_LDS
    async_cp_b32(&sx[0][p], sppT + b * NBINS + p);
    async_cp_b32(&sx[1][p], sppV + ((size_t)(b * CELLS + c)) * NBINS + p);
    async_cp_b32(&sx[2][p], sppI + b * NBINS + p);
    async_cp_b32(&stm[p],   sppTm + ((size_t)(b * 48 + (c % 48))) * NBINS + p);
#else
    sx[0][p] = sppT[b * NBINS + p];
    sx[1][p] = sppV[((size_t)(b * CELLS + c)) * NBINS + p];
    sx[2][p] = sppI[b * NBINS + p];
    stm[p]   = sppTm[((size_t)(b * 48 + (c % 48))) * NBINS + p];
#endif
  }
#if HAVE_ASYNC_LDS
  async_wait0();
#endif
  __syncthreads();
  if (p < NBINS) {
    // branch 3 first conv: (5,3,3), pad 1
    for (int q = 0; q < 5; ++q) {
      float acc = b3a[q];
      for (int j = 0; j < 3; ++j) {
        int pos = p - 1 + j;
        if (pos >= 0 && pos < NBINS)
          for (int ch = 0; ch < 3; ++ch)
            acc += w3a[(q * 3 + ch) * 3 + j] * sx[ch][pos];
      }
      sh[q][p] = relu_f(acc);
    }
  }
  __syncthreads();
  if (p < NBINS) {
    // branch 1: (1,3,1), pad 0
    float a1 = b1[0];
    for (int ch = 0; ch < 3; ++ch) a1 += w1[ch] * sx[ch][p];
    a1 = relu_f(a1);
    // branch 2: (1,3,5), pad 2
    float a2 = b2[0];
    for (int j = 0; j < 5; ++j) {
      int pos = p - 2 + j;
      if (pos >= 0 && pos < NBINS)
        for (int ch = 0; ch < 3; ++ch)
          a2 += w2[ch * 5 + j] * sx[ch][pos];
    }
    a2 = relu_f(a2);
    // branch 3 second conv: (1,5,5), pad 2
    float a3 = b3b[0];
    for (int j = 0; j < 5; ++j) {
      int pos = p - 2 + j;
      if (pos >= 0 && pos < NBINS)
        for (int q = 0; q < 5; ++q)
          a3 += w3b[q * 5 + j] * sh[q][pos];
    }
    a3 = relu_f(a3);
    float y  = relu_f(c1w[0] * a1 + c1w[1] * a2 + c1w[2] * a3 + c1b[0]);
    float zv = relu_f(c2w[0] * y + c2w[1] * stm[p] + c2b[0]);
    Z[(size_t)n * KPAD + p] = zv;
  } else {
    Z[(size_t)n * KPAD + p] = 0.f;   // K padding for WMMA
  }
}

// ---------------------------------------------------------------------------
// Kernel 4: H1 = relu(Z(3072x96) @ Bpad(96x64) + b1) via V_WMMA_F32_16X16X4_F32.
// One wave per 16x16 tile; A (16x96) and B (96x16) tiles DMA'd to LDS by the
// Tensor Data Mover, then the K loop reads fragments from LDS.
// grid = (192, 4), block = 32.
// ---------------------------------------------------------------------------
__global__ void k_gemm_ff1(const float* __restrict__ Z, const float* __restrict__ Bp,
                           const float* __restrict__ bias, float* __restrict__ H1) {
  __shared__ float sA[16][KPAD];   // 16 rows x 96 k   (6 KB)
  __shared__ float sB[KPAD][16];   // 96 k  x 16 cols  (6 KB)
  const int m0 = blockIdx.x * 16;
  const int n0 = blockIdx.y * 16;
  const int l  = threadIdx.x;      // 0..31
  const int ln = l & 15;
  const int kh = l >> 4;

#if HAVE_TDM
  // A: 16 rows of 96 contiguous floats, row stride 96
  tdm_load_2d(lds_off(&sA[0][0]), Z + (size_t)m0 * KPAD,
              /*tensor_d0=*/KPAD, /*tensor_d1=*/16, /*stride0=*/KPAD,
              /*tile_d0=*/KPAD, /*tile_d1=*/16);
  // B: 96 rows of 16 floats starting at column n0, row stride 64
  tdm_load_2d(lds_off(&sB[0][0]), Bp + n0,
              /*tensor_d0=*/H1DIM, /*tensor_d1=*/KPAD, /*stride0=*/H1DIM,
              /*tile_d0=*/16, /*tile_d1=*/KPAD);
  __builtin_amdgcn_s_wait_tensorcnt(0);
  asm volatile("" ::: "memory");
#else
  for (int idx = l; idx < 16 * KPAD; idx += 32)
    sA[idx / KPAD][idx % KPAD] = Z[(size_t)(m0 + idx / KPAD) * KPAD + (idx % KPAD)];
  for (int idx = l; idx < KPAD * 16; idx += 32)
    sB[idx / 16][idx % 16] = Bp[(size_t)(idx / 16) * H1DIM + n0 + (idx % 16)];
  __syncthreads();
#endif

  v8f acc = {};
  for (int k0 = 0; k0 < KPAD; k0 += 4) {
    // A frag: lanes 0-15 hold {K=k0,k0+1}, lanes 16-31 hold {K=k0+2,k0+3}
    v2f a = *(const v2f*)&sA[ln][k0 + 2 * kh];
    v2f bf;
    bf.x = sB[k0 + 2 * kh][ln];
    bf.y = sB[k0 + 2 * kh + 1][ln];
    acc = __builtin_amdgcn_wmma_f32_16x16x4_f32(
        /*neg_a=*/false, a, /*neg_b=*/false, bf,
        /*c_mod=*/(short)0, acc, /*reuse_a=*/false, /*reuse_b=*/false);
  }
  const float bv = bias[n0 + ln];
  for (int v = 0; v < 8; ++v) {
    int row = m0 + v + 8 * kh;     // D layout: vgpr v -> M=v (lanes 0-15) / v+8
    H1[(size_t)row * H1DIM + n0 + ln] = relu_f(acc[v] + bv);
  }
}

// ---------------------------------------------------------------------------
// Kernel 5: feedforward tail per n: 64 -> 16 -> 4 -> 1 (linear+1x1conv, relu)
// ---------------------------------------------------------------------------
__global__ void k_ff_rest(const float* __restrict__ H1,
                          const float* __restrict__ l1w, const float* __restrict__ l1b,
                          const float* __restrict__ c1w, const float* __restrict__ c1b,
                          const float* __restrict__ l2w, const float* __restrict__ l2b,
                          const float* __restrict__ c2w, const float* __restrict__ c2b,
                          const float* __restrict__ l3w, const float* __restrict__ l3b,
                          const float* __restrict__ c3w, const float* __restrict__ c3b,
                          float* __restrict__ Y) {
  int n = blockIdx.x * blockDim.x + threadIdx.x;
  if (n >= NTOT) return;
  float h[64];
  for (int i = 0; i < 64; ++i) h[i] = H1[(size_t)n * H1DIM + i];
  float g[16];
  for (int o = 0; o < 16; ++o) {
    float s1 = l1b[o], s2 = c1b[o];
    for (int i = 0; i < 64; ++i) { float hv = h[i]; s1 += l1w[o * 64 + i] * hv; s2 += c1w[o * 64 + i] * hv; }
    g[o] = relu_f(relu_f(s1) + s2);
  }
  float g2[4];
  for (int o = 0; o < 4; ++o) {
    float s1 = l2b[o], s2 = c2b[o];
    for (int i = 0; i < 16; ++i) { float hv = g[i]; s1 += l2w[o * 16 + i] * hv; s2 += c2w[o * 16 + i] * hv; }
    g2[o] = relu_f(relu_f(s1) + s2);
  }
  float s1 = l3b[0], s2 = c3b[0];
  for (int i = 0; i < 4; ++i) { s1 += l3w[i] * g2[i]; s2 += c3w[i] * g2[i]; }
  Y[n] = relu_f(relu_f(s1) + s2);
}

// ---------------------------------------------------------------------------
// Kernel 6: inception_b on (32, 96) -> (32,). One wave, thread per batch row.
// ---------------------------------------------------------------------------
__global__ void k_incb(const float* __restrict__ Y,
                       const float* __restrict__ w11, const float* __restrict__ bb11,
                       const float* __restrict__ w12, const float* __restrict__ bb12,
                       const float* __restrict__ w21, const float* __restrict__ bb21,
                       const float* __restrict__ w22, const float* __restrict__ bb22,
                       const float* __restrict__ w31, const float* __restrict__ bb31,
                       const float* __restrict__ w32, const float* __restrict__ bb32,
                       const float* __restrict__ w4,  const float* __restrict__ bb4,
                       float* __restrict__ out) {
  int b = threadIdx.x;
  if (b >= BATCH) return;
  float x[96];
  for (int i = 0; i < 96; ++i) x[i] = Y[b * CELLS + i];
  float h[48], r[48];
  for (int o = 0; o < 48; ++o) {
    float s = bb11[o];
    for (int i = 0; i < 96; ++i) s += w11[o * 96 + i] * x[i];
    h[o] = relu_f(s);
  }
  for (int o = 0; o < 48; ++o) {
    float s = bb12[o];
    for (int i = 0; i < 48; ++i) s += w12[o * 48 + i] * h[i];
    r[o] = relu_f(s);
  }
  for (int o = 0; o < 48; ++o) h[o] = relu_f(r[o] + h[o]);
  float h2[16], r2[16];
  for (int o = 0; o < 16; ++o) {
    float s = bb21[o];
    for (int i = 0; i < 48; ++i) s += w21[o * 48 + i] * h[i];
    h2[o] = relu_f(s);
  }
  for (int o = 0; o < 16; ++o) {
    float s = bb22[o];
    for (int i = 0; i < 16; ++i) s += w22[o * 16 + i] * h2[i];
    r2[o] = relu_f(s);
  }
  for (int o = 0; o < 16; ++o) h2[o] = relu_f(r2[o] + h2[o]);
  float h3[4], r3[4];
  for (int o = 0; o < 4; ++o) {
    float s = bb31[o];
    for (int i = 0; i < 16; ++i) s += w31[o * 16 + i] * h2[i];
    h3[o] = relu_f(s);
  }
  for (int o = 0; o < 4; ++o) {
    float s = bb32[o];
    for (int i = 0; i < 4; ++i) s += w32[o * 4 + i] * h3[i];
    r3[o] = relu_f(s);
  }
  for (int o = 0; o < 4; ++o) h3[o] = relu_f(r3[o] + h3[o]);
  float s = bb4[0];
  for (int i = 0; i < 4; ++i) s += w4[i] * h3[i];
  out[b] = relu_f(s);
}

// ---------------------------------------------------------------------------
// Launcher. Input order (setup_inputs dict order, params flattened in
// insertion order, (w, b) pairs):
//   0:T 1:voltages 2:temps 3:currents
//   4,5: fw_lin1 | 6..11: fw_lins[0..2] | 12..17: fw_convs[0..2]
//   18,19:a_b1 20,21:a_b2 22,23:a_b3a 24,25:a_b3b
//   26,27:b_l11 28,29:b_l21 30,31:b_l31 32,33:b_l4
//   34,35:b_l12 36,37:b_l22 38,39:b_l32 40,41:cnn1 42,43:cnn2
// ---------------------------------------------------------------------------
extern "C" void kernel_launch(void* const* d_in, const int* in_sizes, int n_in,
                              void* d_out, int out_size, void* d_ws, size_t ws_size,
                              hipStream_t stream) {
  const float* T  = (const float*)d_in[0];
  const float* V  = (const float*)d_in[1];
  const float* Tm = (const float*)d_in[2];
  const float* I  = (const float*)d_in[3];
  const float* P[44];
  for (int i = 4; i < 44 && i < n_in; ++i) P[i] = (const float*)d_in[i];

  // workspace layout (floats)
  float* ws = (float*)d_ws;
  float* sppT  = ws;                                  // 32*91
  float* sppI  = sppT  + BATCH * NBINS;               // 32*91
  float* sppV  = sppI  + BATCH * NBINS;               // 32*96*91
  float* sppTm = sppV  + BATCH * CELLS * NBINS;       // 32*48*91
  float* Z     = sppTm + BATCH * 48 * NBINS;          // 3072*96
  float* Bpad  = Z     + (size_t)NTOT * KPAD;         // 96*64
  float* H1    = Bpad  + KPAD * H1DIM;                // 3072*64
  float* Y     = H1    + (size_t)NTOT * H1DIM;        // 3072
  (void)ws_size; (void)in_sizes; (void)out_size;

  // Stage 1: SPP
  k_spp<<<dim3(NBINS, BATCH, 2), CELLS, 0, stream>>>(T, V, Tm, I, sppT, sppI, sppV, sppTm);

  // Stage 2: pack B for GEMM (fw_lin1 = P[4], P[5])
  k_prep_b<<<(KPAD * H1DIM + 255) / 256, 256, 0, stream>>>(P[4], Bpad);

  // Stage 3: inception_a + cnn1 + cnn2
  k_incept_a<<<NTOT, CELLS, 0, stream>>>(sppT, sppI, sppV, sppTm,
                                         P[18], P[19],   // a_b1
                                         P[20], P[21],   // a_b2
                                         P[22], P[23],   // a_b3a
                                         P[24], P[25],   // a_b3b
                                         P[40], P[41],   // cnn1
                                         P[42], P[43],   // cnn2
                                         Z);

  // Stage 4: WMMA GEMM  H1 = relu(Z @ W1^T + b1)
  k_gemm_ff1<<<dim3(NTOT / 16, H1DIM / 16), 32, 0, stream>>>(Z, Bpad, P[5], H1);

  // Stage 5: feedforward tail
  k_ff_rest<<<(NTOT + 127) / 128, 128, 0, stream>>>(H1,
                                                    P[6],  P[7],  P[12], P[13],  // 64->16
                                                    P[8],  P[9],  P[14], P[15],  // 16->4
                                                    P[10], P[11], P[16], P[17],  // 4->1
                                                    Y);

  // Stage 6: inception_b -> output (32 floats)
  k_incb<<<1, 32, 0, stream>>>(Y,
                               P[26], P[27], P[34], P[35],   // b_l11, b_l12
                               P[28], P[29], P[36], P[37],   // b_l21, b_l22
                               P[30], P[31], P[38], P[39],   // b_l31, b_l32
                               P[32], P[33],                 // b_l4
                               (float*)d_out);
}